// NTM_17832704213571
// MI455X (gfx1250) — compile-verified
//
#include <hip/hip_runtime.h>
#include <hip/hip_bf16.h>

// ---------------------------------------------------------------------------
// NTM single step, dead-code-eliminated (only `o` is returned, so the whole
// write-head / new_memory path is skipped).
// B=256 NI=512 NO=512 H=1024 N=2048 M=256
// Perf model: 2 streaming passes over 512MB `memory` (~43us @ 23.3TB/s floor);
// GEMMs (~4.3 GFLOP) via v_wmma_f32_16x16x4_f32 (f32-exact, plenty fast here).
// ---------------------------------------------------------------------------

typedef __attribute__((ext_vector_type(2))) float v2f;
typedef __attribute__((ext_vector_type(4))) float v4f;
typedef __attribute__((ext_vector_type(8))) float v8f;

#define Bq   256
#define NIq  512
#define NOq  512
#define Hq   1024
#define Nq   2048
#define Mq   256
#define G4H  4096   // 4*H
#define KIH  768    // NI+M
#define ROW_RO 262  // M+6
#define KOUT 1280   // H+M

__device__ __forceinline__ float sigf(float x)  { return 1.0f / (1.0f + __expf(-x)); }
__device__ __forceinline__ float softplusf(float x) { return (x > 20.0f) ? x : log1pf(expf(x)); }

__device__ __forceinline__ float wave_sum(float v) {
  #pragma unroll
  for (int o = 16; o > 0; o >>= 1) v += __shfl_xor(v, o, 32);
  return v;
}

// =====================  1) gates GEMM (WMMA f32 16x16x4)  ====================
// gates[b, n] = sum_k inp[b,k] W_ih[n,k] + sum_k h0[b,k] W_hh[n,k] + b_ih[n]+b_hh[n]
// inp = [x | init_r] (256 x 768)
__global__ void k_gemm_gates(const float* __restrict__ x, const float* __restrict__ init_r,
                             const float* __restrict__ h0,
                             const float* __restrict__ W_ih, const float* __restrict__ W_hh,
                             const float* __restrict__ b_ih, const float* __restrict__ b_hh,
                             float* __restrict__ gates) {
  const int lane = threadIdx.x & 31, wave = threadIdx.x >> 5;
  const int n0 = blockIdx.x * 128 + wave * 16;   // output-neuron tile
  const int m0 = blockIdx.y * 16;                // batch tile
  const int arow = m0 + (lane & 15);
  const int wrow = n0 + (lane & 15);
  const int koff = (lane >> 4) << 1;             // K sub-offset per ISA A/B layout

  const float* ax = x      + (size_t)arow * NIq;
  const float* ar = init_r + (size_t)arow * Mq;
  const float* ah = h0     + (size_t)arow * Hq;
  const float* wi = W_ih   + (size_t)wrow * KIH;
  const float* wh = W_hh   + (size_t)wrow * Hq;

  v8f c = {};
  #pragma unroll 4
  for (int k = 0; k < KIH; k += 4) {
    int kk = k + koff;
    v2f a, b;
    a.x = (kk     < NIq) ? ax[kk]     : ar[kk - NIq];
    a.y = (kk + 1 < NIq) ? ax[kk + 1] : ar[kk + 1 - NIq];
    b.x = wi[kk]; b.y = wi[kk + 1];
    c = __builtin_amdgcn_wmma_f32_16x16x4_f32(false, a, false, b, (short)0, c, false, false);
  }
  #pragma unroll 4
  for (int k = 0; k < Hq; k += 4) {
    int kk = k + koff;
    v2f a, b;
    a.x = ah[kk]; a.y = ah[kk + 1];
    b.x = wh[kk]; b.y = wh[kk + 1];
    c = __builtin_amdgcn_wmma_f32_16x16x4_f32(false, a, false, b, (short)0, c, false, false);
  }

  const int col = n0 + (lane & 15);
  const int rb  = m0 + ((lane >> 4) << 3);
  const float bias = b_ih[col] + b_hh[col];
  #pragma unroll
  for (int j = 0; j < 8; ++j)
    gates[(size_t)(rb + j) * G4H + col] = c[j] + bias;
}

// =====================  2) LSTM pointwise -> ctrl  ===========================
__global__ void k_lstm(const float* __restrict__ gates, const float* __restrict__ c0,
                       float* __restrict__ ctrl) {
  int idx = blockIdx.x * blockDim.x + threadIdx.x;   // B*H
  int b = idx >> 10, j = idx & (Hq - 1);
  const float* g = gates + (size_t)b * G4H;
  float ig = g[j], fg = g[Hq + j], gg = g[2 * Hq + j], og = g[3 * Hq + j];
  float c  = sigf(fg) * c0[idx] + sigf(ig) * tanhf(gg);
  ctrl[idx] = sigf(og) * tanhf(c);
}

// =====================  3) ro = ctrl @ fc_read_w^T + b  ======================
__global__ void k_gemm_ro(const float* __restrict__ ctrl, const float* __restrict__ W,
                          const float* __restrict__ bias, float* __restrict__ ro) {
  const int lane = threadIdx.x & 31, wave = threadIdx.x >> 5;
  const int tile = blockIdx.x * 8 + wave;
  if (tile >= 17) return;                         // ceil(262/16), wave-uniform skip
  const int n0 = tile * 16;
  const int m0 = blockIdx.y * 16;
  const int arow  = m0 + (lane & 15);
  const int wrow  = min(n0 + (lane & 15), ROW_RO - 1);   // clamp OOB weight rows
  const int koff  = (lane >> 4) << 1;

  const float* ac = ctrl + (size_t)arow * Hq;
  const float* wp = W    + (size_t)wrow * Hq;

  v8f c = {};
  #pragma unroll 4
  for (int k = 0; k < Hq; k += 4) {
    int kk = k + koff;
    v2f a, b;
    a.x = ac[kk]; a.y = ac[kk + 1];
    b.x = wp[kk]; b.y = wp[kk + 1];
    c = __builtin_amdgcn_wmma_f32_16x16x4_f32(false, a, false, b, (short)0, c, false, false);
  }
  const int col = n0 + (lane & 15);
  if (col < ROW_RO) {
    const int rb = m0 + ((lane >> 4) << 3);
    float bv = bias[col];
    #pragma unroll
    for (int j = 0; j < 8; ++j)
      ro[(size_t)(rb + j) * ROW_RO + col] = c[j] + bv;
  }
}

// =====================  4) k=tanh(Kr), ||k||, head params  ===================
// params[b][0..5] = beta, g, s0, s1, s2, L
__global__ void k_prep(const float* __restrict__ ro, float* __restrict__ kbuf,
                       float* __restrict__ kn, float* __restrict__ params) {
  __shared__ float red[256];
  int b = blockIdx.x, t = threadIdx.x;
  const float* r = ro + (size_t)b * ROW_RO;
  float v = tanhf(r[t]);
  kbuf[(size_t)b * Mq + t] = v;
  red[t] = v * v;
  __syncthreads();
  for (int s = 128; s > 0; s >>= 1) { if (t < s) red[t] += red[t + s]; __syncthreads(); }
  if (t == 0) {
    kn[b] = fmaxf(sqrtf(red[0]), 1e-8f);
    float beta = softplusf(r[Mq]);
    float g    = sigf(r[Mq + 1]);
    float a0 = r[Mq + 2], a1 = r[Mq + 3], a2 = r[Mq + 4];
    float mx = fmaxf(a0, fmaxf(a1, a2));
    float e0 = expf(a0 - mx), e1 = expf(a1 - mx), e2 = expf(a2 - mx);
    float se = e0 + e1 + e2;
    float L = 1.0f + softplusf(r[Mq + 5]);
    float* p = params + b * 6;
    p[0] = beta; p[1] = g; p[2] = e0 / se; p[3] = e1 / se; p[4] = e2 / se; p[5] = L;
  }
}

// ======  5) streaming pass 1: num[b,n]=k.mem_row, mn2[b,n]=||mem_row||^2  ====
__global__ void k_mem_pass1(const float* __restrict__ mem, const float* __restrict__ kbuf,
                            float* __restrict__ num, float* __restrict__ mn2) {
  const int b = blockIdx.y;
  const int lane = threadIdx.x & 31, wave = threadIdx.x >> 5;
  const float* kb = kbuf + (size_t)b * Mq;
  v4f k0 = *(const v4f*)(kb + lane * 4);
  v4f k1 = *(const v4f*)(kb + 128 + lane * 4);
  const int row0 = blockIdx.x * 64 + wave * 8;
  #pragma unroll
  for (int i = 0; i < 8; ++i) {
    int row = row0 + i;
    const float* mr = mem + ((size_t)b * Nq + row) * Mq;
    v4f m0 = __builtin_nontemporal_load((const v4f*)(mr + lane * 4));
    v4f m1 = __builtin_nontemporal_load((const v4f*)(mr + 128 + lane * 4));
    float d = m0.x*k0.x + m0.y*k0.y + m0.z*k0.z + m0.w*k0.w
            + m1.x*k1.x + m1.y*k1.y + m1.z*k1.z + m1.w*k1.w;
    float q = m0.x*m0.x + m0.y*m0.y + m0.z*m0.z + m0.w*m0.w
            + m1.x*m1.x + m1.y*m1.y + m1.z*m1.z + m1.w*m1.w;
    d = wave_sum(d); q = wave_sum(q);
    if (lane == 0) { num[(size_t)b * Nq + row] = d; mn2[(size_t)b * Nq + row] = q; }
  }
}

// ======  6) addressing: softmax, interpolate, shift, sharpen, renorm  ========
__global__ void k_address(const float* __restrict__ num, const float* __restrict__ mn2,
                          const float* __restrict__ kn, const float* __restrict__ params,
                          const float* __restrict__ wprev, float* __restrict__ wr) {
  __shared__ float wg[Nq];
  __shared__ float red[256];
  const int b = blockIdx.x, t = threadIdx.x;
  const float* p = params + b * 6;
  const float beta = p[0], g = p[1], sh0 = p[2], sh1 = p[3], sh2 = p[4], L = p[5];
  const float knb = kn[b];
  const size_t base = (size_t)b * Nq;

  float sc[8];
  float lmax = -1e30f;
  #pragma unroll
  for (int j = 0; j < 8; ++j) {
    int n = t + j * 256;
    float mn = fmaxf(sqrtf(mn2[base + n]), 1e-8f);
    sc[j] = beta * num[base + n] / (knb * mn);
    lmax = fmaxf(lmax, sc[j]);
  }
  red[t] = lmax; __syncthreads();
  for (int s = 128; s > 0; s >>= 1) { if (t < s) red[t] = fmaxf(red[t], red[t + s]); __syncthreads(); }
  const float gmax = red[0]; __syncthreads();

  float lsum = 0.0f;
  #pragma unroll
  for (int j = 0; j < 8; ++j) { sc[j] = expf(sc[j] - gmax); lsum += sc[j]; }
  red[t] = lsum; __syncthreads();
  for (int s = 128; s > 0; s >>= 1) { if (t < s) red[t] += red[t + s]; __syncthreads(); }
  const float gsum = red[0]; __syncthreads();

  #pragma unroll
  for (int j = 0; j < 8; ++j) {
    int n = t + j * 256;
    float wc = sc[j] / gsum;
    wg[n] = g * wc + (1.0f - g) * wprev[base + n];
  }
  __syncthreads();

  float pw[8];
  float lsum2 = 0.0f;
  #pragma unroll
  for (int j = 0; j < 8; ++j) {
    int n = t + j * 256;
    float v = sh0 * wg[(n + Nq - 1) & (Nq - 1)] + sh1 * wg[n] + sh2 * wg[(n + 1) & (Nq - 1)];
    v = powf(v, L);
    pw[j] = v; lsum2 += v;
  }
  red[t] = lsum2; __syncthreads();
  for (int s = 128; s > 0; s >>= 1) { if (t < s) red[t] += red[t + s]; __syncthreads(); }
  const float norm = red[0] + 1e-16f;
  #pragma unroll
  for (int j = 0; j < 8; ++j) wr[base + t + j * 256] = pw[j] / norm;
}

// ======  7) streaming pass 2: rpart[b,chunk,m] = sum_n w[b,n] mem[b,n,m]  ====
__global__ void k_mem_pass2(const float* __restrict__ mem, const float* __restrict__ wr,
                            float* __restrict__ rpart) {
  __shared__ float w[256];
  const int ch = blockIdx.x, b = blockIdx.y, t = threadIdx.x;
  w[t] = wr[(size_t)b * Nq + ch * 256 + t];
  __syncthreads();
  const float* base = mem + ((size_t)b * Nq + (size_t)ch * 256) * Mq + t;
  float a0 = 0.f, a1 = 0.f, a2 = 0.f, a3 = 0.f;
  for (int i = 0; i < 256; i += 4) {
    a0 += w[i]     * __builtin_nontemporal_load(base + (size_t)(i)     * Mq);
    a1 += w[i + 1] * __builtin_nontemporal_load(base + (size_t)(i + 1) * Mq);
    a2 += w[i + 2] * __builtin_nontemporal_load(base + (size_t)(i + 2) * Mq);
    a3 += w[i + 3] * __builtin_nontemporal_load(base + (size_t)(i + 3) * Mq);
  }
  rpart[((size_t)b * 8 + ch) * Mq + t] = (a0 + a1) + (a2 + a3);
}

// =====================  8) reduce 8 chunks -> r[b,m]  ========================
__global__ void k_reduce_r(const float* __restrict__ rpart, float* __restrict__ r) {
  const int b = blockIdx.x, m = threadIdx.x;
  const float* p = rpart + (size_t)b * 8 * Mq + m;
  float s = 0.f;
  #pragma unroll
  for (int c = 0; c < 8; ++c) s += p[c * Mq];
  r[(size_t)b * Mq + m] = s;
}

// =====  9) o = sigmoid([ctrl|r] @ fc_out_w^T + b)  (WMMA f32 16x16x4)  =======
__global__ void k_gemm_out(const float* __restrict__ ctrl, const float* __restrict__ r,
                           const float* __restrict__ W, const float* __restrict__ bias,
                           float* __restrict__ out) {
  const int lane = threadIdx.x & 31, wave = threadIdx.x >> 5;
  const int n0 = (blockIdx.x * 8 + wave) * 16;
  const int m0 = blockIdx.y * 16;
  const int arow = m0 + (lane & 15);
  const int wrow = n0 + (lane & 15);
  const int koff = (lane >> 4) << 1;

  const float* ac = ctrl + (size_t)arow * Hq;
  const float* ar = r    + (size_t)arow * Mq;
  const float* wp = W    + (size_t)wrow * KOUT;

  v8f c = {};
  #pragma unroll 4
  for (int k = 0; k < KOUT; k += 4) {
    int kk = k + koff;
    v2f a, b;
    a.x = (kk     < Hq) ? ac[kk]     : ar[kk - Hq];
    a.y = (kk + 1 < Hq) ? ac[kk + 1] : ar[kk + 1 - Hq];
    b.x = wp[kk]; b.y = wp[kk + 1];
    c = __builtin_amdgcn_wmma_f32_16x16x4_f32(false, a, false, b, (short)0, c, false, false);
  }
  const int col = n0 + (lane & 15);
  const int rb  = m0 + ((lane >> 4) << 3);
  const float bv = bias[col];
  #pragma unroll
  for (int j = 0; j < 8; ++j)
    out[(size_t)(rb + j) * NOq + col] = sigf(c[j] + bv);
}

// ============================================================================
extern "C" void kernel_launch(void* const* d_in, const int* in_sizes, int n_in,
                              void* d_out, int out_size, void* d_ws, size_t ws_size,
                              hipStream_t stream) {
  const float* x         = (const float*)d_in[0];
  const float* memory    = (const float*)d_in[1];
  const float* W_ih      = (const float*)d_in[2];
  const float* W_hh      = (const float*)d_in[3];
  const float* b_ih      = (const float*)d_in[4];
  const float* b_hh      = (const float*)d_in[5];
  const float* h0        = (const float*)d_in[6];
  const float* c0        = (const float*)d_in[7];
  const float* fc_read_w = (const float*)d_in[8];
  const float* fc_read_b = (const float*)d_in[9];
  // d_in[10], d_in[11]: fc_write_* -- dead (output `o` does not depend on them)
  const float* fc_out_w  = (const float*)d_in[12];
  const float* fc_out_b  = (const float*)d_in[13];
  const float* init_r    = (const float*)d_in[14];
  const float* w_prev_r  = (const float*)d_in[15];
  float* out = (float*)d_out;

  float* p = (float*)d_ws;
  float* gates  = p; p += (size_t)Bq * G4H;   // 1,048,576
  float* ctrl   = p; p += (size_t)Bq * Hq;    //   262,144
  float* ro     = p; p += (size_t)Bq * ROW_RO;//    67,072
  float* kbuf   = p; p += (size_t)Bq * Mq;    //    65,536
  float* knb    = p; p += Bq;                 //       256
  float* params = p; p += Bq * 6;             //     1,536
  float* numb   = p; p += (size_t)Bq * Nq;    //   524,288
  float* mn2    = p; p += (size_t)Bq * Nq;    //   524,288
  float* wr     = p; p += (size_t)Bq * Nq;    //   524,288
  float* rpart  = p; p += (size_t)Bq * 8 * Mq;//   524,288
  float* rbuf   = p; p += (size_t)Bq * Mq;    //    65,536   (~14.4 MB total)

  k_gemm_gates<<<dim3(32, 16), 256, 0, stream>>>(x, init_r, h0, W_ih, W_hh, b_ih, b_hh, gates);
  k_lstm      <<<dim3(1024),   256, 0, stream>>>(gates, c0, ctrl);
  k_gemm_ro   <<<dim3(3, 16),  256, 0, stream>>>(ctrl, fc_read_w, fc_read_b, ro);
  k_prep      <<<dim3(Bq),     256, 0, stream>>>(ro, kbuf, knb, params);
  k_mem_pass1 <<<dim3(32, Bq), 256, 0, stream>>>(memory, kbuf, numb, mn2);
  k_address   <<<dim3(Bq),     256, 0, stream>>>(numb, mn2, knb, params, w_prev_r, wr);
  k_mem_pass2 <<<dim3(8, Bq),  256, 0, stream>>>(memory, wr, rpart);
  k_reduce_r  <<<dim3(Bq),     256, 0, stream>>>(rpart, rbuf);
  k_gemm_out  <<<dim3(4, 16),  256, 0, stream>>>(ctrl, rbuf, fc_out_w, fc_out_b, out);
}